// reaction_gcn_90872918049163
// MI455X (gfx1250) — compile-verified
//
#include <hip/hip_runtime.h>
#include <hip/hip_bf16.h>
#include <math.h>

// Reference shapes
#define N_NODES   307
#define N_EDGES   340
#define L_DIM     64
#define KPAD      308        // 77 * 4, K padded for 16x16x4 WMMA steps
#define ROW_TILES 20         // 20*16 = 320 >= 307
#define C_CH      2

typedef __attribute__((ext_vector_type(2))) float v2f;
typedef __attribute__((ext_vector_type(8))) float v8f;

// One workgroup per batch element. 128 threads = 4 wave32; wave w owns output
// columns [16w, 16w+16). x (307x64 f32) is staged once into LDS; the Laplacian
// is built sparsely per 16-row tile in LDS (ds_add_f32 atomics) and consumed
// by full-precision f32 WMMA (v_wmma_f32_16x16x4_f32).
__launch_bounds__(128)
__global__ void reaction_gcn_kernel(const float* __restrict__ inputs,
                                    const int*   __restrict__ ind,
                                    const int*   __restrict__ edge_from,
                                    const int*   __restrict__ edge_to,
                                    const float* __restrict__ weight_react,
                                    const float* __restrict__ bias_reaction,
                                    float* __restrict__ out)
{
    __shared__ float x_lds[KPAD * L_DIM];    // 308*64*4  = 78848 B (row 307 zero)
    __shared__ float lap[16 * KPAD];         // 16*308*4  = 19712 B
    __shared__ float w_lds[N_EDGES];
    __shared__ int   ef_lds[N_EDGES];
    __shared__ int   et_lds[N_EDGES];
    __shared__ float bias_lds[ROW_TILES * 16];

    const int b    = blockIdx.x;
    const int tid  = threadIdx.x;      // 0..127
    const int lane = tid & 31;
    const int wave = tid >> 5;         // 0..3 -> column tile of L

    const int bucket = ind[b];         // RESOLUTION == 1

    // ---- stage x[b] = inputs[b, 0, :, :] into LDS (coalesced, read once) ----
    const float* xg = inputs + (size_t)b * (C_CH * N_NODES * L_DIM);
    for (int i = tid; i < N_NODES * L_DIM; i += 128) x_lds[i] = xg[i];
    for (int i = tid; i < L_DIM; i += 128) x_lds[N_NODES * L_DIM + i] = 0.0f; // K pad row

    // ---- stage edge list, bucket weights, bias ----
    for (int e = tid; e < N_EDGES; e += 128) {
        ef_lds[e] = edge_from[e];
        et_lds[e] = edge_to[e];
        w_lds[e]  = weight_react[(size_t)bucket * N_EDGES + e];
    }
    for (int v = tid; v < ROW_TILES * 16; v += 128)
        bias_lds[v] = (v < N_NODES) ? bias_reaction[(size_t)bucket * N_NODES + v] : 0.0f;

    __syncthreads();

    float* outb = out + (size_t)b * (N_NODES * L_DIM);

    for (int rt = 0; rt < ROW_TILES; ++rt) {
        const int r0 = rt * 16;

        // ---- build 16x308 Laplacian row tile sparsely in LDS ----
        for (int i = tid; i < 16 * KPAD; i += 128) lap[i] = 0.0f;
        __syncthreads();
        for (int e = tid; e < N_EDGES; e += 128) {
            const int   f = ef_lds[e];
            const int   t = et_lds[e];
            const float w = w_lds[e];
            if (f >= r0 && f < r0 + 16) {          // row f: Lap[f][t]-=w, Lap[f][f]+=w
                atomicAdd(&lap[(f - r0) * KPAD + t], -w);
                atomicAdd(&lap[(f - r0) * KPAD + f],  w);
            }
            if (t >= r0 && t < r0 + 16) {          // row t: Lap[t][f]-=w, Lap[t][t]+=w
                atomicAdd(&lap[(t - r0) * KPAD + f], -w);
                atomicAdd(&lap[(t - r0) * KPAD + t],  w);
            }
        }
        __syncthreads();

        // ---- WMMA: acc(16x16) = lap_tile(16x308) @ x(308, cols of this wave) ----
        const int m     = lane & 15;             // A: row M; B/C/D: col N
        const int khalf = (lane >> 4) << 1;      // 0 (lanes 0-15) or 2 (lanes 16-31)
        const int col   = wave * 16 + m;         // global L column

        v8f acc = {};
        for (int k0 = 0; k0 < KPAD; k0 += 4) {
            v2f afrag, bfrag;
            afrag.x = lap[m * KPAD + (k0 + khalf)];
            afrag.y = lap[m * KPAD + (k0 + khalf + 1)];
            bfrag.x = x_lds[(k0 + khalf)     * L_DIM + col];
            bfrag.y = x_lds[(k0 + khalf + 1) * L_DIM + col];
            // 8 args: (neg_a, A, neg_b, B, c_mod, C, reuse_a, reuse_b)
            acc = __builtin_amdgcn_wmma_f32_16x16x4_f32(
                false, afrag, false, bfrag, (short)0, acc, false, false);
        }

        // ---- epilogue: out = tanh(acc + bias) + x ----
        const int rowbase = r0 + ((lane >> 4) << 3);   // lanes>=16 hold M = g+8
        #pragma unroll
        for (int g = 0; g < 8; ++g) {
            const int row = rowbase + g;
            if (row < N_NODES) {
                const float r = acc[g] + bias_lds[row];
                outb[row * L_DIM + col] = tanhf(r) + x_lds[row * L_DIM + col];
            }
        }
        __syncthreads();   // lap/x reuse barrier before next row tile
    }
}

extern "C" void kernel_launch(void* const* d_in, const int* in_sizes, int n_in,
                              void* d_out, int out_size, void* d_ws, size_t ws_size,
                              hipStream_t stream)
{
    (void)in_sizes; (void)n_in; (void)out_size; (void)d_ws; (void)ws_size;
    const float* inputs        = (const float*)d_in[0];   // (256,2,307,64) f32
    const int*   ind           = (const int*)  d_in[1];   // (256,)
    const int*   edge_from     = (const int*)  d_in[2];   // (340,)
    const int*   edge_to       = (const int*)  d_in[3];   // (340,)
    const float* weight_react  = (const float*)d_in[4];   // (288,340) f32
    const float* bias_reaction = (const float*)d_in[5];   // (288,307) f32
    float*       out           = (float*)d_out;           // (256,307,64) f32

    reaction_gcn_kernel<<<256, 128, 0, stream>>>(
        inputs, ind, edge_from, edge_to, weight_react, bias_reaction, out);
}